// Network_33320356282492
// MI455X (gfx1250) — compile-verified
//
#include <hip/hip_runtime.h>
#include <stdint.h>

// Problem dims (fixed by the reference)
#define NB 16
#define NC 3
#define NH 512
#define NW 512
#define HWSZ (NH * NW)
#define CHW  (NC * HWSZ)
#define NFEAT 120

#define WAVES 8                 // waves per block, one row per wave
#define NTHREADS (WAVES * 32)   // wave32
#define NITER 4                 // float4 quads per lane: 4 * 4 * 32 = 512 = NW

// ---- CDNA5 async global->LDS DMA (inline asm; assembled by MC, so rc==0
// proves the gfx1250 async path is emitted)
__device__ __forceinline__ void async_ld_b128(uint32_t lds_off, const float* g) {
  // VDST = per-lane LDS byte offset, VADDR = per-lane 64-bit global address
  asm volatile("global_load_async_to_lds_b128 %0, %1, off"
               :: "v"(lds_off), "v"(g) : "memory");
}
#define S_WAIT_ASYNC(n) asm volatile("s_wait_asynccnt " #n ::: "memory")
#define S_WAIT_DSCNT0() asm volatile("s_wait_dscnt 0" ::: "memory")

__global__ __launch_bounds__(NTHREADS)
void fused_lightconv_poly_kernel(const float* __restrict__ x,
                                 const float* __restrict__ feat,
                                 float* __restrict__ out)
{
  // per-wave private double buffer: [wave][buf][channel][lane] float4 = 24 KB/block
  __shared__ float4 tile[WAVES][2][NC][32];

  const int tid  = threadIdx.x;
  const int wave = tid >> 5;
  const int lane = tid & 31;
  const int row  = blockIdx.x * WAVES + wave;   // [0, NB*NH)
  // batch is uniform per block (8 rows/block, 512 rows/batch) -> scalar loads
  const int b    = (int)(blockIdx.x >> 6);
  const int h    = row & (NH - 1);

  const float* fb = feat + b * NFEAT;

  // ---------------------------------------------------------------------
  // Phase 1: fold the 20-term polynomial by powers of x and hoist the
  // h-dependent parts. Do this FIRST: the 60 poly coefficients are only
  // transiently live (they die into 30 per-wave VGPR values, h is wave-
  // divergent), so they don't overlap the 60 conv params in the SGPR file.
  //   att = A0(w) + A1(w)*x + A2(w)*x^2 + A3*x^3  per channel
  // ---------------------------------------------------------------------
  const float hn  = (float)h * (2.0f / (float)(NH - 1)) - 1.0f;
  const float hh  = hn * hn, hhh = hh * hn;

  float a00[3], a01[3], a02[3], a03[3];   // A0 = a00 + a01 w + a02 w^2 + a03 w^3
  float a10[3], a11[3], a12[3];           // A1 = a10 + a11 w + a12 w^2
  float a20[3], a21[3];                   // A2 = a20 + a21 w
  float a30[3];                           // A3
#pragma unroll
  for (int c = 0; c < 3; ++c) {
#define CO(j) fb[60 + (j) * 3 + c]
    a00[c] = fmaf(CO(10), hhh, fmaf(CO(4), hh, fmaf(CO(1), hn, CO(0))));
    a01[c] = fmaf(CO(13), hh, fmaf(CO(7), hn, CO(2)));
    a02[c] = fmaf(CO(16), hn, CO(5));
    a03[c] = CO(11);
    a10[c] = fmaf(CO(14), hh, fmaf(CO(8), hn, CO(3)));
    a11[c] = fmaf(CO(19), hn, CO(9));
    a12[c] = CO(15);
    a20[c] = fmaf(CO(17), hn, CO(6));
    a21[c] = CO(18);
    a30[c] = CO(12);
#undef CO
  }

  // ---------------------------------------------------------------------
  // Phase 2: conv params (kernel 3x3, bias 3, slope 3, for each of 4 convs)
  // — 60 uniforms, stays within the SGPR file now.
  // ---------------------------------------------------------------------
  float kc[4][9], bi[4][3], sl[4][3];
#pragma unroll
  for (int i = 0; i < 4; ++i) {
#pragma unroll
    for (int k = 0; k < 9; ++k) kc[i][k] = fb[i * 15 + k];
#pragma unroll
    for (int c = 0; c < 3; ++c) {
      bi[i][c] = fb[i * 15 + 9 + c];
      sl[i][c] = fb[i * 15 + 12 + c];
    }
  }

  const float* xrow = x   + (size_t)b * CHW + (size_t)h * NW;
  float*       orow = out + (size_t)b * CHW + (size_t)h * NW;

  // low 32 bits of a generic shared pointer == LDS byte offset (aperture in hi32)
  const uint32_t lds0 = (uint32_t)(uintptr_t)(&tile[wave][0][0][0]);

  auto issue = [&](int j) {
    const int buf = j & 1;
#pragma unroll
    for (int c = 0; c < NC; ++c) {
      async_ld_b128(lds0 + (uint32_t)(((buf * NC + c) * 32 + lane) * 16),
                    xrow + c * HWSZ + j * 128 + lane * 4);
    }
  };

  // depth-2 double-buffered async prefetch pipeline (ASYNCcnt is per-wave,
  // completion is in-order, and each wave consumes only its own loads ->
  // no barriers needed)
  issue(0);
  issue(1);

#pragma unroll
  for (int j = 0; j < NITER; ++j) {
    // 3 async loads per iteration; waiting to <=3 means iteration j's loads
    // have landed. Last iteration drains fully.
    if (j < NITER - 1) { S_WAIT_ASYNC(3); } else { S_WAIT_ASYNC(0); }

    const int buf = j & 1;
    float4 p0 = tile[wave][buf][0][lane];
    float4 p1 = tile[wave][buf][1][lane];
    float4 p2 = tile[wave][buf][2][lane];

    // LDS reads must reach VGPRs before the j+2 prefetch overwrites this buffer
    S_WAIT_DSCNT0();
    if (j + 2 < NITER) issue(j + 2);

    float xv[3][4] = { {p0.x, p0.y, p0.z, p0.w},
                       {p1.x, p1.y, p1.z, p1.w},
                       {p2.x, p2.y, p2.z, p2.w} };
    float rv[3][4];
    const float wbase = (float)(j * 128 + lane * 4);

#pragma unroll
    for (int k = 0; k < 4; ++k) {
      float x0 = xv[0][k], x1 = xv[1][k], x2 = xv[2][k];

      // 4 chained light convs: y = x + (K x + b); branch-free PReLU
#pragma unroll
      for (int i = 0; i < 4; ++i) {
        const float i0 = x0, i1 = x1, i2 = x2;
        float y0 = fmaf(kc[i][0], i0, fmaf(kc[i][1], i1, fmaf(kc[i][2], i2, i0 + bi[i][0])));
        float y1 = fmaf(kc[i][3], i0, fmaf(kc[i][4], i1, fmaf(kc[i][5], i2, i1 + bi[i][1])));
        float y2 = fmaf(kc[i][6], i0, fmaf(kc[i][7], i1, fmaf(kc[i][8], i2, i2 + bi[i][2])));
        x0 = fmaf(sl[i][0], fminf(y0, 0.0f), fmaxf(y0, 0.0f));
        x1 = fmaf(sl[i][1], fminf(y1, 0.0f), fmaxf(y1, 0.0f));
        x2 = fmaf(sl[i][2], fminf(y2, 0.0f), fmaxf(y2, 0.0f));
      }

      // polynomial attention: Horner in w, then Horner in x
      const float wn = fmaf(wbase + (float)k, 2.0f / (float)(NW - 1), -1.0f);
#pragma unroll
      for (int c = 0; c < 3; ++c) {
        const float xc = (c == 0) ? x0 : (c == 1) ? x1 : x2;
        const float A0 = fmaf(fmaf(fmaf(a03[c], wn, a02[c]), wn, a01[c]), wn, a00[c]);
        const float A1 = fmaf(fmaf(a12[c], wn, a11[c]), wn, a10[c]);
        const float A2 = fmaf(a21[c], wn, a20[c]);
        const float att = fmaf(fmaf(fmaf(a30[c], xc, A2), xc, A1), xc, A0);
        rv[c][k] = fmaf(att, xc, xc);   // att*x + x
      }
    }

#pragma unroll
    for (int c = 0; c < NC; ++c) {
      *(float4*)(orow + c * HWSZ + j * 128 + lane * 4) =
          make_float4(rv[c][0], rv[c][1], rv[c][2], rv[c][3]);
    }
  }
}

extern "C" void kernel_launch(void* const* d_in, const int* in_sizes, int n_in,
                              void* d_out, int out_size, void* d_ws, size_t ws_size,
                              hipStream_t stream) {
  const float* x    = (const float*)d_in[0];   // (16, 3, 512, 512) f32
  const float* feat = (const float*)d_in[1];   // (16, 120) f32
  float* out = (float*)d_out;                  // (16, 3, 512, 512) f32

  const int rows = NB * NH;                    // 8192 rows, 8 per block
  fused_lightconv_poly_kernel<<<dim3(rows / WAVES), dim3(NTHREADS), 0, stream>>>(x, feat, out);

  (void)in_sizes; (void)n_in; (void)out_size; (void)d_ws; (void)ws_size;
}